// SaltAndPepper_26792005992691
// MI455X (gfx1250) — compile-verified
//
#include <hip/hip_runtime.h>
#include <cstdint>
#include <cstddef>

// ---------------------------------------------------------------------------
// Salt & pepper noise, bit-exact JAX threefry2x32 reproduction.
//   image : (3, 4096, 4096) f32
//   r     : jax.random.uniform(key(42), (4096,4096))
//   out   : r < 0.005 ? 1.0 : (r < 0.01 ? 0.0 : image)   (r broadcast over c)
//
// Memory-bound: 402 MB traffic -> ~17us @ 23.3 TB/s. PRNG regenerated in VALU
// (threefry per pair computed ONCE, reused for all 3 channels). All traffic is
// b128, non-temporal (streaming working set > 192MB L2).
// ---------------------------------------------------------------------------

typedef float fvec4 __attribute__((ext_vector_type(4)));

#define PLANE      16777216u   // 4096*4096 = 2^24 elements per channel
#define HALF_PAIRS 8388608u    // 2^23 threefry pairs
#define PAIRS_PER_THREAD 4u

__device__ __forceinline__ uint32_t rotl32(uint32_t x, uint32_t r) {
  return (x << r) | (x >> (32u - r));
}

// JAX threefry2x32, key = (0, 42), 20 rounds.
__device__ __forceinline__ void threefry2x32_k42(uint32_t x0, uint32_t x1,
                                                 uint32_t& o0, uint32_t& o1) {
  const uint32_t ks0 = 0u;
  const uint32_t ks1 = 42u;
  const uint32_t ks2 = ks0 ^ ks1 ^ 0x1BD11BDAu;

  x0 += ks0; x1 += ks1;
#define TF_ROUND(r) { x0 += x1; x1 = rotl32(x1, (r)); x1 ^= x0; }
  TF_ROUND(13u) TF_ROUND(15u) TF_ROUND(26u) TF_ROUND(6u)
  x0 += ks1; x1 += ks2 + 1u;
  TF_ROUND(17u) TF_ROUND(29u) TF_ROUND(16u) TF_ROUND(24u)
  x0 += ks2; x1 += ks0 + 2u;
  TF_ROUND(13u) TF_ROUND(15u) TF_ROUND(26u) TF_ROUND(6u)
  x0 += ks0; x1 += ks1 + 3u;
  TF_ROUND(17u) TF_ROUND(29u) TF_ROUND(16u) TF_ROUND(24u)
  x0 += ks1; x1 += ks2 + 4u;
  TF_ROUND(13u) TF_ROUND(15u) TF_ROUND(26u) TF_ROUND(6u)
  x0 += ks2; x1 += ks0 + 5u;
#undef TF_ROUND
  o0 = x0; o1 = x1;
}

// JAX uniform [0,1): bitcast((bits>>9) | 0x3F800000) - 1.0f
__device__ __forceinline__ float bits_to_unit(uint32_t b) {
  return __uint_as_float((b >> 9) | 0x3F800000u) - 1.0f;
}

__device__ __forceinline__ float sp_select(float r, float v) {
  // r < 0.005 -> SALT(1.0); else r < 0.01 -> PEPPER(0.0); else image
  return (r < 0.005f) ? 1.0f : ((r < 0.01f) ? 0.0f : v);
}

__global__ __launch_bounds__(256)
void SaltAndPepper_26792005992691_kernel(const float* __restrict__ img,
                                         float* __restrict__ out) {
  const uint32_t t = blockIdx.x * blockDim.x + threadIdx.x; // 0 .. 2^21-1
  const uint32_t p = t * PAIRS_PER_THREAD;                  // pair base (elem idx)

  // Generate 4 threefry pairs -> 8 uniform values (4 low-half, 4 high-half).
  float rlo[4], rhi[4];
#pragma unroll
  for (uint32_t k = 0; k < 4u; ++k) {
    uint32_t y0, y1;
    threefry2x32_k42(p + k, p + k + HALF_PAIRS, y0, y1);
    rlo[k] = bits_to_unit(y0);
    rhi[k] = bits_to_unit(y1);
  }

  // Apply to all 3 channels (same noise plane), b128 non-temporal streaming.
  const fvec4* __restrict__ src = (const fvec4*)img;
  fvec4* __restrict__ dst = (fvec4*)out;
  const uint32_t v4_plane = PLANE / 4u;       // 4194304 float4 per channel
  const uint32_t v4_half  = HALF_PAIRS / 4u;  // 2097152
#pragma unroll
  for (uint32_t c = 0; c < 3u; ++c) {
    const size_t iLo = (size_t)c * v4_plane + t;
    const size_t iHi = iLo + v4_half;

    fvec4 vLo = __builtin_nontemporal_load(src + iLo);
    fvec4 vHi = __builtin_nontemporal_load(src + iHi);

    fvec4 wLo, wHi;
#pragma unroll
    for (uint32_t k = 0; k < 4u; ++k) {
      wLo[k] = sp_select(rlo[k], vLo[k]);
      wHi[k] = sp_select(rhi[k], vHi[k]);
    }
    __builtin_nontemporal_store(wLo, dst + iLo);
    __builtin_nontemporal_store(wHi, dst + iHi);
  }
}

extern "C" void kernel_launch(void* const* d_in, const int* in_sizes, int n_in,
                              void* d_out, int out_size, void* d_ws, size_t ws_size,
                              hipStream_t stream) {
  (void)in_sizes; (void)n_in; (void)out_size; (void)d_ws; (void)ws_size;
  const float* img = (const float*)d_in[0];
  float* out = (float*)d_out;

  // 2^23 pairs / 4 per thread = 2^21 threads = 8192 blocks x 256 threads
  // (8 wave32 per block; 64K waves -> saturates all HBM channels).
  const dim3 block(256);
  const dim3 grid(8192);
  SaltAndPepper_26792005992691_kernel<<<grid, block, 0, stream>>>(img, out);
}